// ExplicitSheafLaplacian_26173530701948
// MI455X (gfx1250) — compile-verified
//
#include <hip/hip_runtime.h>
#include <hip/hip_bf16.h>

// ---------------------------------------------------------------------------
// ExplicitSheafLaplacian coboundary energy, CDNA5 (gfx1250), f32 WMMA path.
//
//   proj[v]   = data[v] @ proj_w + proj_b                 (N x 256 @ 256 x 32)
//   gate(ctx) = sigmoid(relu(ctx @ w1 + b1) @ w2 + b2)    (N x 5 -> N x 32)
//   energy    = sum_e sum_{n,k} (proj[src]*g_src - proj[tgt]*g_tgt)^2
//               / (6 * N * 32)
//
// One wave32 owns a 16-node tile. Both GEMMs use V_WMMA_F32_16X16X4_F32.
// ---------------------------------------------------------------------------

typedef float v2f __attribute__((ext_vector_type(2)));
typedef float v8f __attribute__((ext_vector_type(8)));

#define NV_   4
#define NN_   131072
#define DSTK_ 256
#define DEDG_ 32
#define NCTX_ 5
#define NTILE (NN_ / 16)
#define WTPAD 260   // padded LDS stride: 260 % 64 == 4 -> conflict-free b64 reads
#define NBLK  256
#define NTHR  256

__device__ __forceinline__ v8f wmma_f32_16x16x4(v2f a, v2f b, v8f c) {
  // 8-arg form: (neg_a, A, neg_b, B, c_mod, C, reuse_a, reuse_b)
  return __builtin_amdgcn_wmma_f32_16x16x4_f32(false, a, false, b, (short)0, c,
                                               false, false);
}

__launch_bounds__(NTHR, 1)
__global__ void sheaf_energy_kernel(const float* __restrict__ data,
                                    const float* __restrict__ ctx_src,
                                    const float* __restrict__ ctx_tgt,
                                    const float* __restrict__ proj_w,
                                    const float* __restrict__ proj_b,
                                    const float* __restrict__ gate_w1,
                                    const float* __restrict__ gate_b1,
                                    const float* __restrict__ gate_w2,
                                    const float* __restrict__ gate_b2,
                                    float* __restrict__ partials) {
  __shared__ float wt[32 * WTPAD];   // proj_w transposed, padded (33280 B)
  __shared__ float red[NTHR];

  const int tid  = threadIdx.x;
  const int lane = tid & 31;
  const int ll   = lane & 15;   // column / node-in-tile index
  const int hi   = lane >> 4;   // K-half selector for A/B tiles

  // ---- stage proj_w transposed into LDS: wt[c][k] = proj_w[k][c] ----------
  for (int i = tid; i < DSTK_ * DEDG_; i += NTHR) {
    const int kk = i >> 5, c = i & 31;
    wt[c * WTPAD + kk] = proj_w[i];
  }
  __syncthreads();

  // ---- per-lane gate weights ----------------------------------------------
  // Layer-1 output h is produced directly in WMMA A-layout: for k-step s,
  // lane holds hidden indices j = 4*s + 2*hi + {0,1}.
  float w1L[NCTX_][8];
  float b1L[8];
  v2f   w2B[8];   // [s][t]: B-tile of w2 (4x16) for k-step s, col-tile t
#pragma unroll
  for (int s = 0; s < 4; ++s) {
#pragma unroll
    for (int p = 0; p < 2; ++p) {
      const int j = 4 * s + 2 * hi + p;
      b1L[s * 2 + p] = gate_b1[j];
#pragma unroll
      for (int c = 0; c < NCTX_; ++c) w1L[c][s * 2 + p] = gate_w1[c * 16 + j];
    }
#pragma unroll
    for (int t = 0; t < 2; ++t) {
      const int kb = 4 * s + 2 * hi;
      v2f b;
      b.x = gate_w2[(kb + 0) * DEDG_ + t * 16 + ll];
      b.y = gate_w2[(kb + 1) * DEDG_ + t * 16 + ll];
      w2B[s * 2 + t] = b;
    }
  }
  const float pbv[2] = {proj_b[ll], proj_b[16 + ll]};
  const float b2v[2] = {gate_b2[ll], gate_b2[16 + ll]};

  // gate(ctx row block) -> two 16x16 f32 tiles in C/D layout
  auto compute_gate = [&](const float* __restrict__ cb, v8f& g0, v8f& g1) {
    float cc[NCTX_];
#pragma unroll
    for (int q = 0; q < NCTX_; ++q) cc[q] = cb[q];
    v2f H[4];
#pragma unroll
    for (int s = 0; s < 4; ++s) {
      float h0 = b1L[s * 2 + 0];
      float h1 = b1L[s * 2 + 1];
#pragma unroll
      for (int q = 0; q < NCTX_; ++q) {
        h0 = fmaf(cc[q], w1L[q][s * 2 + 0], h0);
        h1 = fmaf(cc[q], w1L[q][s * 2 + 1], h1);
      }
      H[s].x = fmaxf(h0, 0.0f);
      H[s].y = fmaxf(h1, 0.0f);
    }
#pragma unroll
    for (int r = 0; r < 8; ++r) { g0[r] = b2v[0]; g1[r] = b2v[1]; }
#pragma unroll
    for (int s = 0; s < 4; ++s) {
      g0 = wmma_f32_16x16x4(H[s], w2B[s * 2 + 0], g0);
      g1 = wmma_f32_16x16x4(H[s], w2B[s * 2 + 1], g1);
    }
#pragma unroll
    for (int r = 0; r < 8; ++r) {
      g0[r] = 1.0f / (1.0f + __expf(-g0[r]));
      g1[r] = 1.0f / (1.0f + __expf(-g1[r]));
    }
  };

  // ---- tile loop (wave-uniform bounds -> EXEC stays all-ones for WMMA) ----
  const int    wpb   = NTHR >> 5;
  const int    gw    = blockIdx.x * wpb + (tid >> 5);
  const int    nw    = gridDim.x * wpb;
  const size_t laneA = (size_t)ll * DSTK_ + 2 * hi;   // A-layout lane offset
  const size_t vstep = (size_t)NN_ * DSTK_;

  float acc = 0.0f;

  for (int tile = gw; tile < NTILE; tile += nw) {
    const int n0 = tile << 4;

    // prefetch next tile's streaming data (global_prefetch_b8)
    const int tn = tile + nw;
    if (tn < NTILE) {
      const float* pf = data + (size_t)(tn << 4) * DSTK_ + laneA;
#pragma unroll
      for (int v = 0; v < NV_; ++v) __builtin_prefetch(pf + v * vstep, 0, 1);
    }

    // ---- projection: 4 views x (N=16 tile) x 32 cols, K=256 --------------
    v8f P[NV_][2];
#pragma unroll
    for (int v = 0; v < NV_; ++v)
#pragma unroll
      for (int t = 0; t < 2; ++t)
#pragma unroll
        for (int r = 0; r < 8; ++r) P[v][t][r] = pbv[t];

    const float* dbase = data + (size_t)n0 * DSTK_ + laneA;
#pragma unroll 4
    for (int k = 0; k < DSTK_; k += 4) {
      const int kb = k + 2 * hi;
      const v2f B0 = *reinterpret_cast<const v2f*>(&wt[ll * WTPAD + kb]);
      const v2f B1 = *reinterpret_cast<const v2f*>(&wt[(16 + ll) * WTPAD + kb]);
#pragma unroll
      for (int v = 0; v < NV_; ++v) {
        const v2f A = *reinterpret_cast<const v2f*>(dbase + v * vstep + k);
        P[v][0] = wmma_f32_16x16x4(A, B0, P[v][0]);
        P[v][1] = wmma_f32_16x16x4(A, B1, P[v][1]);
      }
    }

    // ---- edges: gates via WMMA, then elementwise energy ------------------
#pragma unroll
    for (int e = 0; e < 6; ++e) {
      constexpr int ESRC[6] = {0, 0, 0, 1, 1, 2};
      constexpr int ETGT[6] = {1, 2, 3, 2, 3, 3};
      const float* cs = ctx_src + ((size_t)e * NN_ + n0 + ll) * NCTX_;
      const float* ct = ctx_tgt + ((size_t)e * NN_ + n0 + ll) * NCTX_;
      v8f Gs0, Gs1, Gt0, Gt1;
      compute_gate(cs, Gs0, Gs1);
      compute_gate(ct, Gt0, Gt1);
      const int si = ESRC[e], ti = ETGT[e];
#pragma unroll
      for (int r = 0; r < 8; ++r) {
        const float d0 = P[si][0][r] * Gs0[r] - P[ti][0][r] * Gt0[r];
        const float d1 = P[si][1][r] * Gs1[r] - P[ti][1][r] * Gt1[r];
        acc = fmaf(d0, d0, acc);
        acc = fmaf(d1, d1, acc);
      }
    }
  }

  // ---- deterministic block reduction -> per-block partial -----------------
  red[tid] = acc;
  __syncthreads();
  for (int s = NTHR >> 1; s > 0; s >>= 1) {
    if (tid < s) red[tid] += red[tid + s];
    __syncthreads();
  }
  if (tid == 0) partials[blockIdx.x] = red[0];
}

__global__ void sheaf_reduce_kernel(const float* __restrict__ partials, int n,
                                    float* __restrict__ out) {
  __shared__ float red[NTHR];
  float s = 0.0f;
  for (int i = threadIdx.x; i < n; i += NTHR) s += partials[i];
  red[threadIdx.x] = s;
  __syncthreads();
  for (int k = NTHR >> 1; k > 0; k >>= 1) {
    if (threadIdx.x < k) red[threadIdx.x] += red[threadIdx.x + k];
    __syncthreads();
  }
  if (threadIdx.x == 0)
    out[0] = red[0] * (1.0f / (6.0f * (float)NN_ * (float)DEDG_));
}

extern "C" void kernel_launch(void* const* d_in, const int* in_sizes, int n_in,
                              void* d_out, int out_size, void* d_ws,
                              size_t ws_size, hipStream_t stream) {
  (void)in_sizes; (void)n_in; (void)out_size; (void)ws_size;
  const float* data    = (const float*)d_in[0];
  const float* ctx_src = (const float*)d_in[1];
  const float* ctx_tgt = (const float*)d_in[2];
  const float* proj_w  = (const float*)d_in[3];
  const float* proj_b  = (const float*)d_in[4];
  const float* gate_w1 = (const float*)d_in[5];
  const float* gate_b1 = (const float*)d_in[6];
  const float* gate_w2 = (const float*)d_in[7];
  const float* gate_b2 = (const float*)d_in[8];
  float* partials = (float*)d_ws;
  float* out      = (float*)d_out;

  sheaf_energy_kernel<<<NBLK, NTHR, 0, stream>>>(
      data, ctx_src, ctx_tgt, proj_w, proj_b, gate_w1, gate_b1, gate_w2,
      gate_b2, partials);
  sheaf_reduce_kernel<<<1, NTHR, 0, stream>>>(partials, NBLK, out);
}